// SpikingGatingNetwork_54881092108731
// MI455X (gfx1250) — compile-verified
//
#include <hip/hip_runtime.h>
#include <hip/hip_bf16.h>

// ---- problem constants (match reference) ----
#define B_      8
#define LV_     1024
#define LA_     512
#define LI_     512
#define D_      1024
#define T_      2048            // LV+LA+LI
#define H1_     512             // D/2
#define H2_     256             // D/4
#define TARGET_ 614             // int(0.3 * T)
#define DECAY_  0.95f
#define LEAK_   0.1f
#define THRESH_ 1.0f

// padded LDS strides (stride % 64 == 4 -> conflict-free column reads)
#define KC_   128               // K-chunk for X staging
#define XCS_  132               // 128 + 4-dword pad (matches TDM pad descriptor)
#define H1S_  516
#define H2S_  260

typedef __attribute__((ext_vector_type(2))) float        v2f;
typedef __attribute__((ext_vector_type(8))) float        v8f;
typedef __attribute__((ext_vector_type(4))) unsigned int u32x4;
typedef __attribute__((ext_vector_type(4))) int          i32x4;
typedef __attribute__((ext_vector_type(8))) int          i32x8;

#if defined(__gfx1250__) && __has_builtin(__builtin_amdgcn_tensor_load_to_lds)
#define USE_TDM 1
#endif

__device__ __forceinline__ v8f wmma_f32_16x16x4(v2f a, v2f b, v8f c) {
  // D = A(16x4,f32) * B(4x16,f32) + C(16x16,f32)
  return __builtin_amdgcn_wmma_f32_16x16x4_f32(false, a, false, b, (short)0, c,
                                               false, false);
}

#ifdef USE_TDM
// TDM: DMA a 16x128 f32 tile (row stride D_ elements) from global into LDS,
// inserting a 4-dword pad after every 128-dword row => LDS row stride 132.
__device__ __forceinline__ void tdm_load_tile16x128(const float* gsrc,
                                                    unsigned lds_byte_off) {
  const unsigned long long ga = (unsigned long long)(size_t)gsrc;
  u32x4 g0;
  g0.x = 1u;                                            // count=1 (valid user D#)
  g0.y = lds_byte_off;                                  // lds_addr (bytes)
  g0.z = (unsigned)ga;                                  // global_addr[31:0]
  g0.w = ((unsigned)(ga >> 32) & 0x01ffffffu)           // global_addr[56:32]
         | 0x80000000u;                                 // type=2 ("image")
  i32x8 g1;
  g1[0] = (int)((2u << 16)        // data_size = 4 bytes
                | (1u << 20)      // pad_enable
                | (6u << 22)      // pad_interval: 128 dwords
                | (3u << 25));    // pad_amount : 4 dwords
  g1[1] = 0;                      // barrier addr 0 | tensor_dim0[15:0]=0
  g1[2] = 0x00004000;             // tensor_dim0 = 1<<30 (no clip) | tdim1 lo = 0
  g1[3] = (int)((128u << 16) | 0x00004000u); // tile_dim0=128 | tensor_dim1=1<<30
  g1[4] = 16;                     // tile_dim1 = 16, tile_dim2 = 0
  g1[5] = D_;                     // tensor_dim0_stride = 1024 elements
  g1[6] = 0;
  g1[7] = 0;
  const i32x4 gz = {0, 0, 0, 0};
#if __clang_major__ >= 23
  const i32x8 gz8 = {0, 0, 0, 0, 0, 0, 0, 0};
  __builtin_amdgcn_tensor_load_to_lds(g0, g1, gz, gz, gz8, 0);
#else
  __builtin_amdgcn_tensor_load_to_lds(g0, g1, gz, gz, 0);
#endif
}
#endif

__device__ __forceinline__ void coop_stage16x128(const float* src, int kc,
                                                 float* dstbuf, int tid) {
  for (int i = tid; i < 16 * (KC_ / 4); i += 256) {
    const int row = i >> 5;
    const int c4  = i & 31;
    const float4 tv = reinterpret_cast<const float4*>(src + (size_t)row * D_ + kc)[c4];
    float* dst = dstbuf + row * XCS_ + c4 * 4;
    dst[0] = tv.x; dst[1] = tv.y; dst[2] = tv.z; dst[3] = tv.w;
  }
}

// =====================================================================
// Kernel 1: fused gate MLP -> membrane potential per token.
//   16 token rows / block, 8 waves. X chunks double-buffered in LDS and
//   (when available) fetched by the Tensor Data Mover overlapped with the
//   V_WMMA_F32_16X16X4_F32 main loops. Embeds added at A-frag read time.
// =====================================================================
__global__ __launch_bounds__(256)
void sgn_mlp_kernel(const float* __restrict__ vis, const float* __restrict__ aud,
                    const float* __restrict__ imu, const float* __restrict__ mem_in,
                    const float* __restrict__ ev,  const float* __restrict__ ea,
                    const float* __restrict__ ei,
                    const float* __restrict__ W1, const float* __restrict__ b1,
                    const float* __restrict__ W2, const float* __restrict__ b2,
                    const float* __restrict__ W3, const float* __restrict__ b3,
                    float* __restrict__ mem_out)
{
  __shared__ float sXc[2][16 * XCS_];  // 2 x 8.25 KB double-buffered X chunk
  __shared__ float sEmb[D_];           // 4 KB modality embed
  __shared__ float sH1[16 * H1S_];     // 32.25 KB relu(X@W1+b1)
  __shared__ float sH2[16 * H2S_];     // 16.25 KB relu(H1@W2+b2)

  const int tid  = threadIdx.x;
  const int lane = tid & 31;
  const int wave = tid >> 5;
  const int r0   = blockIdx.x * 16;        // first global token row of tile
  const int b    = r0 / T_;
  const int t0   = r0 % T_;                // tile never crosses modality/batch

  const float* src; const float* emb;
  if (t0 < LV_)            { src = vis + ((size_t)b * LV_ + t0) * D_;               emb = ev; }
  else if (t0 < LV_ + LA_) { src = aud + ((size_t)b * LA_ + (t0 - LV_)) * D_;       emb = ea; }
  else                     { src = imu + ((size_t)b * LI_ + (t0 - LV_ - LA_)) * D_; emb = ei; }

  const int m    = lane & 15;
  const int koff = (lane < 16) ? 0 : 2;
  const int ncol = lane & 15;

  const v8f zero8 = {0.f, 0.f, 0.f, 0.f, 0.f, 0.f, 0.f, 0.f};

  // stage embed once (broadcast-read later, conflict-free)
  for (int i = tid; i < D_; i += 256) sEmb[i] = emb[i];

  // ---- prolog: chunk 0 into buffer 0 ----
#ifdef USE_TDM
  const unsigned ldsoff[2] = { (unsigned)(size_t)(void*)&sXc[0][0],
                               (unsigned)(size_t)(void*)&sXc[1][0] };
  if (wave == 0) {
    tdm_load_tile16x128(src, ldsoff[0]);
    __builtin_amdgcn_s_wait_tensorcnt(0);
  }
#else
  coop_stage16x128(src, 0, &sXc[0][0], tid);
#endif
  __syncthreads();

  // ---------- Stage 1: H1 = relu(X @ W1 + b1) ----------
  v8f acc1[4];
  #pragma unroll
  for (int t = 0; t < 4; ++t) acc1[t] = zero8;

  int cur = 0;
  for (int kc = 0; kc < D_; kc += KC_) {
    const int nxt = cur ^ 1;
    if (kc + KC_ < D_) {
#ifdef USE_TDM
      if (wave == 0) tdm_load_tile16x128(src + (kc + KC_), ldsoff[nxt]);
#else
      coop_stage16x128(src, kc + KC_, &sXc[nxt][0], tid);
#endif
    }

    const float* xb = &sXc[cur][0];
    for (int kk = 0; kk < KC_; kk += 4) {
      const int kg = kc + kk + koff;
      v2f a;
      a.x = xb[m * XCS_ + kk + koff]     + sEmb[kg];
      a.y = xb[m * XCS_ + kk + koff + 1] + sEmb[kg + 1];
      #pragma unroll
      for (int t = 0; t < 4; ++t) {
        const int n = (wave * 4 + t) * 16 + ncol;
        v2f bb;
        bb.x = W1[(size_t)kg * H1_ + n];
        bb.y = W1[(size_t)(kg + 1) * H1_ + n];
        acc1[t] = wmma_f32_16x16x4(a, bb, acc1[t]);
      }
    }

#ifdef USE_TDM
    if (wave == 0) __builtin_amdgcn_s_wait_tensorcnt(0);
#endif
    __syncthreads();   // next chunk visible; cur buffer free for reuse
    cur = nxt;
  }

  // epilogue: +b1, relu -> sH1  (C/D layout: VGPR j = row j (+8 for hi lanes))
  #pragma unroll
  for (int t = 0; t < 4; ++t) {
    const int n = (wave * 4 + t) * 16 + ncol;
    const float bias = b1[n];
    #pragma unroll
    for (int j = 0; j < 8; ++j) {
      const int row = j + ((lane < 16) ? 0 : 8);
      const float v = acc1[t][j] + bias;
      sH1[row * H1S_ + n] = v > 0.f ? v : 0.f;
    }
  }
  __syncthreads();

  // ---------- Stage 2: H2 = relu(H1 @ W2 + b2) ----------
  v8f acc2[2];
  acc2[0] = zero8; acc2[1] = zero8;
  for (int k = 0; k < H1_; k += 4) {
    v2f a;
    a.x = sH1[m * H1S_ + k + koff];
    a.y = sH1[m * H1S_ + k + koff + 1];
    #pragma unroll
    for (int t = 0; t < 2; ++t) {
      const int n = (wave * 2 + t) * 16 + ncol;
      v2f bb;
      bb.x = W2[(size_t)(k + koff)     * H2_ + n];
      bb.y = W2[(size_t)(k + koff + 1) * H2_ + n];
      acc2[t] = wmma_f32_16x16x4(a, bb, acc2[t]);
    }
  }
  #pragma unroll
  for (int t = 0; t < 2; ++t) {
    const int n = (wave * 2 + t) * 16 + ncol;
    const float bias = b2[n];
    #pragma unroll
    for (int j = 0; j < 8; ++j) {
      const int row = j + ((lane < 16) ? 0 : 8);
      const float v = acc2[t][j] + bias;
      sH2[row * H2S_ + n] = v > 0.f ? v : 0.f;
    }
  }
  __syncthreads();

  // ---------- Stage 3: importance + leaky-integrate membrane ----------
  if (tid < 16) {
    float s = 0.f;
    for (int k = 0; k < H2_; ++k) s += sH2[tid * H2S_ + k] * W3[k];
    const float imp = s + b3[0];
    const int r = r0 + tid;
    mem_out[r] = (DECAY_ * mem_in[r] + imp) * (1.0f - LEAK_);
  }
}

// =====================================================================
// Kernel 2: per-batch firing + radix top-k selection + pack indices
// =====================================================================
__global__ __launch_bounds__(256)
void sgn_select_kernel(const float* __restrict__ memv,
                       float* __restrict__ sel_out,
                       float* __restrict__ newmem_out,
                       int* __restrict__ srcidx, int* __restrict__ counts)
{
  __shared__ unsigned int sKey[T_];
  __shared__ float        sVal[T_];
  __shared__ int tCnt[256];
  __shared__ int sScan[256];
  __shared__ int sRed;
  __shared__ int sTot;

  const int b   = blockIdx.x;
  const int tid = threadIdx.x;
  const float* mb = memv + (size_t)b * T_;

  // load, fire mask, monotone float->uint key (firing masked to -inf key)
  int fcnt = 0;
  for (int i = tid; i < T_; i += 256) {
    const float v = mb[i];
    sVal[i] = v;
    const bool fire = (v >= THRESH_);
    fcnt += fire ? 1 : 0;
    const unsigned bits = __float_as_uint(fire ? -__builtin_inff() : v);
    sKey[i] = (bits & 0x80000000u) ? ~bits : (bits | 0x80000000u);
  }
  tCnt[tid] = fcnt;
  __syncthreads();
  if (tid == 0) { int s = 0; for (int i = 0; i < 256; ++i) s += tCnt[i]; sTot = s; }
  __syncthreads();
  const int Ftot = sTot;
  int need = TARGET_ - Ftot; if (need < 0) need = 0;

  // bitwise max{x : count(key >= x) >= need}  == need-th largest key
  unsigned Kstar = 0;
  if (need > 0) {
    unsigned prefix = 0;
    for (int bit = 31; bit >= 0; --bit) {
      const unsigned cand = prefix | (1u << bit);
      if (tid == 0) sRed = 0;
      __syncthreads();
      int c = 0;
      for (int i = tid; i < T_; i += 256) c += (sKey[i] >= cand) ? 1 : 0;
      atomicAdd(&sRed, c);
      __syncthreads();
      if (sRed >= need) prefix = cand;
      __syncthreads();
    }
    Kstar = prefix;
  }

  // count strictly-greater keys
  if (tid == 0) sRed = 0;
  __syncthreads();
  if (need > 0) {
    int c = 0;
    for (int i = tid; i < T_; i += 256) c += (sKey[i] > Kstar) ? 1 : 0;
    atomicAdd(&sRed, c);
  }
  __syncthreads();
  const int rem = (need > 0) ? (need - sRed) : 0;

  // exclusive scan of per-thread "key==Kstar" counts (index-ascending ties)
  int eqc = 0;
  #pragma unroll
  for (int q = 0; q < 8; ++q) {
    const int i = tid * 8 + q;
    eqc += (need > 0 && sKey[i] == Kstar) ? 1 : 0;
  }
  tCnt[tid] = eqc;
  __syncthreads();
  if (tid == 0) { int run = 0; for (int i = 0; i < 256; ++i) { sScan[i] = run; run += tCnt[i]; } }
  __syncthreads();
  const int eqbase = sScan[tid];

  // selection decision for this thread's 8 contiguous tokens
  int selflag[8];
  int selcnt = 0, eqseen = 0;
  #pragma unroll
  for (int q = 0; q < 8; ++q) {
    const int i = tid * 8 + q;
    const float v = sVal[i];
    const int fire = (v >= THRESH_) ? 1 : 0;
    int s = fire;
    if (!fire && need > 0) {
      const unsigned k = sKey[i];
      if (k > Kstar) s = 1;
      else if (k == Kstar && (eqbase + eqseen) < rem) s = 1;
    }
    if (need > 0 && sKey[i] == Kstar) eqseen++;
    selflag[q] = s;
    selcnt += s;
  }

  // pack positions (stable, original order)
  tCnt[tid] = selcnt;
  __syncthreads();
  if (tid == 0) {
    int run = 0;
    for (int i = 0; i < 256; ++i) { sScan[i] = run; run += tCnt[i]; }
    counts[b] = run;
  }
  __syncthreads();
  int off = sScan[tid];
  #pragma unroll
  for (int q = 0; q < 8; ++q) {
    const int i = tid * 8 + q;
    const int s = selflag[q];
    sel_out[(size_t)b * T_ + i]    = s ? 1.0f : 0.0f;
    newmem_out[(size_t)b * T_ + i] = s ? 0.0f : sVal[i];
    if (s) { srcidx[(size_t)b * T_ + off] = i; ++off; }
  }
}

// =====================================================================
// Kernel 3: gather selected rows (token + embed), zero-pad the rest
// =====================================================================
__global__ __launch_bounds__(256)
void sgn_pack_kernel(const float* __restrict__ vis, const float* __restrict__ aud,
                     const float* __restrict__ imu,
                     const float* __restrict__ ev, const float* __restrict__ ea,
                     const float* __restrict__ ei,
                     const int* __restrict__ srcidx, const int* __restrict__ counts,
                     float* __restrict__ out)
{
  const int row = blockIdx.x;           // 0 .. B*T-1
  const int b = row / T_, p = row % T_;
  const int tid = threadIdx.x;          // 256 threads, 1 float4 each (D=1024)
  float4* orow = reinterpret_cast<float4*>(out + (size_t)row * D_);

  if (p < counts[b]) {
    const int t = srcidx[(size_t)b * T_ + p];
    const float* src; const float* emb;
    if (t < LV_)            { src = vis + ((size_t)b * LV_ + t) * D_;               emb = ev; }
    else if (t < LV_ + LA_) { src = aud + ((size_t)b * LA_ + (t - LV_)) * D_;       emb = ea; }
    else                    { src = imu + ((size_t)b * LI_ + (t - LV_ - LA_)) * D_; emb = ei; }
    const float4 tv  = reinterpret_cast<const float4*>(src)[tid];
    const float4 evv = reinterpret_cast<const float4*>(emb)[tid];
    orow[tid] = make_float4(tv.x + evv.x, tv.y + evv.y, tv.z + evv.z, tv.w + evv.w);
  } else {
    orow[tid] = make_float4(0.f, 0.f, 0.f, 0.f);
  }
}

// =====================================================================
extern "C" void kernel_launch(void* const* d_in, const int* in_sizes, int n_in,
                              void* d_out, int out_size, void* d_ws, size_t ws_size,
                              hipStream_t stream)
{
  (void)in_sizes; (void)n_in; (void)out_size; (void)ws_size;

  const float* vis    = (const float*)d_in[0];
  const float* aud    = (const float*)d_in[1];
  const float* imu    = (const float*)d_in[2];
  const float* mem_in = (const float*)d_in[3];
  const float* ev     = (const float*)d_in[4];
  const float* ea     = (const float*)d_in[5];
  const float* ei     = (const float*)d_in[6];
  const float* W1     = (const float*)d_in[7];
  const float* b1     = (const float*)d_in[8];
  const float* W2     = (const float*)d_in[9];
  const float* b2     = (const float*)d_in[10];
  const float* W3     = (const float*)d_in[11];
  const float* b3     = (const float*)d_in[12];

  // outputs: padded [B,T,D] | sel [B,T] | new_mem [B,T]  (all f32)
  float* out_padded = (float*)d_out;
  float* out_sel    = out_padded + (size_t)B_ * T_ * D_;
  float* out_newmem = out_sel + (size_t)B_ * T_;

  // workspace: mem f32[B*T] | srcidx i32[B*T] | counts i32[B]
  float* ws_mem = (float*)d_ws;
  int*   ws_src = (int*)((char*)d_ws + (size_t)B_ * T_ * sizeof(float));
  int*   ws_cnt = (int*)((char*)d_ws + 2 * (size_t)B_ * T_ * sizeof(float));

  sgn_mlp_kernel<<<(B_ * T_) / 16, 256, 0, stream>>>(
      vis, aud, imu, mem_in, ev, ea, ei, W1, b1, W2, b2, W3, b3, ws_mem);

  sgn_select_kernel<<<B_, 256, 0, stream>>>(
      ws_mem, out_sel, out_newmem, ws_src, ws_cnt);

  sgn_pack_kernel<<<B_ * T_, 256, 0, stream>>>(
      vis, aud, imu, ev, ea, ei, ws_src, ws_cnt, out_padded);
}